// HyperGNNLayer_42502996361358
// MI455X (gfx1250) — compile-verified
//
#include <hip/hip_runtime.h>

#define B_DIM 8
#define N_DIM 1024
#define F_DIM 64
#define EPS_V 1e-10f

typedef __attribute__((ext_vector_type(2))) float v2f;
typedef __attribute__((ext_vector_type(4))) float v4f;
typedef __attribute__((ext_vector_type(8))) float v8f;

// ---------------------------------------------------------------------------
// Kernel 1: zero the degree array (workspace is poisoned, must init ourselves)
// ---------------------------------------------------------------------------
__global__ void hg_zero_kernel(float* __restrict__ p, int n) {
    int i = blockIdx.x * blockDim.x + threadIdx.x;
    if (i < n) p[i] = 0.0f;
}

// ---------------------------------------------------------------------------
// Kernel 2: degree count per (b, i) head node
// ---------------------------------------------------------------------------
__global__ void hg_degree_kernel(const int* __restrict__ eb,
                                 const int* __restrict__ ei,
                                 float* __restrict__ deg, int nnz) {
    int e = blockIdx.x * blockDim.x + threadIdx.x;
    if (e < nnz) atomicAdd(&deg[eb[e] * N_DIM + ei[e]], 1.0f);
}

// ---------------------------------------------------------------------------
// Kernel 3: both MLPs via V_WMMA_F32_16X16X4_F32.
//   x1  = relu(relu(x@W1+b1)@W2+b2)   -> ws (consumed by edge kernel)
//   out = relu(relu(x@Ws1+bs1)@Ws2+bs2)  (edge kernel accumulates on top)
// Block = 256 threads = 8 waves; each wave computes a 16-row strip.
// Grid = (B*N) / (8*16) = 64 blocks. Full blocks -> EXEC all-ones for WMMA.
// ---------------------------------------------------------------------------
__global__ __launch_bounds__(256) void hg_mlp_wmma_kernel(
    const float* __restrict__ x,
    const float* __restrict__ W1, const float* __restrict__ b1,
    const float* __restrict__ W2, const float* __restrict__ b2,
    const float* __restrict__ Ws1, const float* __restrict__ bs1,
    const float* __restrict__ Ws2, const float* __restrict__ bs2,
    float* __restrict__ x1_out, float* __restrict__ self_out) {

    __shared__ float sW[F_DIM * F_DIM];   // 16 KB, reused for all 4 weight mats
    __shared__ float sB[F_DIM];
    __shared__ float sH[8][16 * F_DIM];   // 32 KB: per-wave hidden strip

    const int tid  = threadIdx.x;
    const int wave = tid >> 5;
    const int lane = tid & 31;
    const int hlf  = lane >> 4;    // lane half selects K-half
    const int mn   = lane & 15;    // M for A-frags, N for B/C/D-frags
    const int rowBase = (blockIdx.x * 8 + wave) * 16;

    // A-fragments of the input strip X[rowBase..rowBase+15][0..63] (32 VGPRs)
    // A layout (16x4 f32): lane L -> M = L%16 ; V0 = A[M][half*2+0], V1 = +1
    v2f aX[16];
#pragma unroll
    for (int kk = 0; kk < 16; ++kk) {
        const float* p = x + (size_t)(rowBase + mn) * F_DIM + kk * 4 + hlf * 2;
        aX[kk][0] = p[0];
        aX[kk][1] = p[1];
    }

    const float* Wfirst[2]  = {W1, Ws1};
    const float* Bfirst[2]  = {b1, bs1};
    const float* Wsecond[2] = {W2, Ws2};
    const float* Bsecond[2] = {b2, bs2};
    float*       Dst[2]     = {x1_out, self_out};

    for (int s = 0; s < 2; ++s) {
        // ---- stage first-layer weights ----
        __syncthreads();
        for (int i = tid; i < F_DIM * F_DIM; i += 256) sW[i] = Wfirst[s][i];
        if (tid < F_DIM) sB[tid] = Bfirst[s][tid];
        __syncthreads();

        // ---- GEMM1: H = relu(X @ W + b), write to per-wave LDS strip ----
#pragma unroll
        for (int nt = 0; nt < 4; ++nt) {
            v8f c;
            float bias = sB[nt * 16 + mn];
#pragma unroll
            for (int r = 0; r < 8; ++r) c[r] = bias;
#pragma unroll
            for (int kk = 0; kk < 16; ++kk) {
                // B layout (4x16 f32): lane -> N = L%16 ; V0 = B[half*2+0][N]
                v2f bF;
                bF[0] = sW[(kk * 4 + hlf * 2 + 0) * F_DIM + nt * 16 + mn];
                bF[1] = sW[(kk * 4 + hlf * 2 + 1) * F_DIM + nt * 16 + mn];
                c = __builtin_amdgcn_wmma_f32_16x16x4_f32(
                        false, aX[kk], false, bF, (short)0, c, false, false);
            }
            // D layout: VGPR r -> M = r + 8*half, N = lane%16
#pragma unroll
            for (int r = 0; r < 8; ++r)
                sH[wave][(r + 8 * hlf) * F_DIM + nt * 16 + mn] = fmaxf(c[r], 0.0f);
        }
        __syncthreads();

        // ---- stage second-layer weights ----
        for (int i = tid; i < F_DIM * F_DIM; i += 256) sW[i] = Wsecond[s][i];
        if (tid < F_DIM) sB[tid] = Bsecond[s][tid];
        __syncthreads();

        // ---- A-fragments of H from LDS ----
        v2f aH[16];
#pragma unroll
        for (int kk = 0; kk < 16; ++kk) {
            aH[kk][0] = sH[wave][mn * F_DIM + kk * 4 + hlf * 2 + 0];
            aH[kk][1] = sH[wave][mn * F_DIM + kk * 4 + hlf * 2 + 1];
        }

        // ---- GEMM2: OUT = relu(H @ W + b), store to global ----
#pragma unroll
        for (int nt = 0; nt < 4; ++nt) {
            v8f c;
            float bias = sB[nt * 16 + mn];
#pragma unroll
            for (int r = 0; r < 8; ++r) c[r] = bias;
#pragma unroll
            for (int kk = 0; kk < 16; ++kk) {
                v2f bF;
                bF[0] = sW[(kk * 4 + hlf * 2 + 0) * F_DIM + nt * 16 + mn];
                bF[1] = sW[(kk * 4 + hlf * 2 + 1) * F_DIM + nt * 16 + mn];
                c = __builtin_amdgcn_wmma_f32_16x16x4_f32(
                        false, aH[kk], false, bF, (short)0, c, false, false);
            }
#pragma unroll
            for (int r = 0; r < 8; ++r) {
                Dst[s][(size_t)(rowBase + r + 8 * hlf) * F_DIM + nt * 16 + mn] =
                    fmaxf(c[r], 0.0f);
            }
        }
    }
}

// ---------------------------------------------------------------------------
// Kernel 4: edge streaming. 16 lanes per edge, float4 per lane (one W_val row
// = 256B = one fully coalesced 16x16B access group).
//
// MI455X cache reasoning: W_val (256MB) is touched exactly once -> load with
// the NT temporal hint so it doesn't contend for L2 ways. The x1 gather source
// (2MB) and the atomic scatter target (2MB) are re-referenced ~64M times and
// must stay L2-resident -> default RT policy. Index arrays are also single-use
// -> NT. HBM roofline: 256MB / 23.3 TB/s ~= 11us for this kernel.
// ---------------------------------------------------------------------------
__global__ __launch_bounds__(256) void hg_edge_kernel(
    const float* __restrict__ Wval,
    const int* __restrict__ eb, const int* __restrict__ ei,
    const int* __restrict__ ej, const int* __restrict__ ek,
    const float* __restrict__ deg, const float* __restrict__ x1,
    float* __restrict__ out, int nnz) {

    int gid = blockIdx.x * blockDim.x + threadIdx.x;
    int e = gid >> 4;
    if (e >= nnz) return;
    int c = (gid & 15) * 4;

    // single-use index data: non-temporal
    int b = __builtin_nontemporal_load(eb + e);
    int i = __builtin_nontemporal_load(ei + e);
    int j = __builtin_nontemporal_load(ej + e);
    int k = __builtin_nontemporal_load(ek + e);

    float a = 1.0f / (deg[b * N_DIM + i] + EPS_V);

    // 256MB one-shot stream: non-temporal so x1/out stay hot in L2
    const v4f w = __builtin_nontemporal_load(
        (const v4f*)(Wval + (size_t)e * F_DIM + c));
    // hot, reused data: regular temporal loads
    const float4 xk = *(const float4*)(x1 + (size_t)(b * N_DIM + k) * F_DIM + c);
    const float4 xj = *(const float4*)(x1 + (size_t)(b * N_DIM + j) * F_DIM + c);

    float* o = out + (size_t)(b * N_DIM + i) * F_DIM + c;
    atomicAdd(o + 0, a * w[0] * xk.x * xj.x);
    atomicAdd(o + 1, a * w[1] * xk.y * xj.y);
    atomicAdd(o + 2, a * w[2] * xk.z * xj.z);
    atomicAdd(o + 3, a * w[3] * xk.w * xj.w);
}

// ---------------------------------------------------------------------------
// Host launcher
// ---------------------------------------------------------------------------
extern "C" void kernel_launch(void* const* d_in, const int* in_sizes, int n_in,
                              void* d_out, int out_size, void* d_ws, size_t ws_size,
                              hipStream_t stream) {
    const float* x    = (const float*)d_in[0];
    const float* Wval = (const float*)d_in[1];
    const int*   eb   = (const int*)d_in[2];
    const int*   ei   = (const int*)d_in[3];
    const int*   ej   = (const int*)d_in[4];
    const int*   ek   = (const int*)d_in[5];
    const float* W1   = (const float*)d_in[6];
    const float* b1   = (const float*)d_in[7];
    const float* W2   = (const float*)d_in[8];
    const float* b2   = (const float*)d_in[9];
    const float* Ws1  = (const float*)d_in[10];
    const float* bs1  = (const float*)d_in[11];
    const float* Ws2  = (const float*)d_in[12];
    const float* bs2  = (const float*)d_in[13];

    const int nnz   = in_sizes[2];
    const int nodes = B_DIM * N_DIM;

    float* deg = (float*)d_ws;              // nodes floats
    float* x1  = deg + nodes;               // nodes * F_DIM floats
    float* out = (float*)d_out;

    // 1) zero degree accumulator
    hg_zero_kernel<<<(nodes + 255) / 256, 256, 0, stream>>>(deg, nodes);

    // 2) degree count
    hg_degree_kernel<<<(nnz + 255) / 256, 256, 0, stream>>>(eb, ei, deg, nnz);

    // 3) both MLPs (WMMA): x1 -> ws, self-MLP -> d_out
    hg_mlp_wmma_kernel<<<nodes / 128, 256, 0, stream>>>(
        x, W1, b1, W2, b2, Ws1, bs1, Ws2, bs2, x1, out);

    // 4) edge stream + scatter-add on top of self-MLP output
    long long threads = (long long)nnz * 16;
    hg_edge_kernel<<<(unsigned)((threads + 255) / 256), 256, 0, stream>>>(
        Wval, eb, ei, ej, ek, deg, x1, out, nnz);
}